// GAT3Layers_56624848830741
// MI455X (gfx1250) — compile-verified
//
#include <hip/hip_runtime.h>

// ---------------------------------------------------------------------------
// GATv2 x3 layers for MI455X (gfx1250, wave32).
//   - Dense projections x@Wl / x@Wr via V_WMMA_F32_16X16X4_F32 (fp32 WMMA).
//   - Edge softmax/aggregation via L2-resident float4 gathers + f32 atomics.
// Constants from the reference: N=100000, E=1000000, C=64, H=4, O=16.
// ---------------------------------------------------------------------------

typedef float v2f __attribute__((ext_vector_type(2)));
typedef float v8f __attribute__((ext_vector_type(8)));

#define CC 64   // channels
#define HH 4    // heads
#define OO 16   // per-head dim
#define NEG_SLOPE 0.2f

// ---- monotone float<->uint encoding for atomicMax on floats ----------------
__device__ __forceinline__ unsigned fenc(float f) {
  unsigned b = __float_as_uint(f);
  return (b & 0x80000000u) ? ~b : (b | 0x80000000u);
}
__device__ __forceinline__ float fdec(unsigned e) {
  return (e & 0x80000000u) ? __uint_as_float(e & 0x7FFFFFFFu)
                           : __uint_as_float(~e);
}

// ---------------------------------------------------------------------------
// WMMA GEMM: Y[M,64] = X[M,64] @ W[64,64], fp32, 16x16x4 WMMA.
// Block = 128 threads = 4 wave32; wave w computes the 16x16 tile at column
// n0 = 16*w for the 16-row block at m0 = 16*blockIdx.x.
// A layout (16x4 f32): lanes 0-15 -> K={0,1}, lanes 16-31 -> K={2,3}, M=lane%16.
// B layout (4x16 f32): lanes 0-15 -> K={0,1}, lanes 16-31 -> K={2,3}, N=lane%16.
// C/D layout: VGPR r holds M = r + 8*(lane/16), N = lane%16.
// ---------------------------------------------------------------------------
__global__ __launch_bounds__(128) void gat_gemm_wmma(
    const float* __restrict__ X, const float* __restrict__ W,
    float* __restrict__ Y, int nrows) {
  const int wave = threadIdx.x >> 5;
  const int lane = threadIdx.x & 31;
  const int m0 = blockIdx.x * 16;
  if (m0 >= nrows) return;
  const int n0 = wave * 16;
  const int lrow = lane & 15;
  const int khalf = (lane >> 4) << 1;   // 0 or 2

  const float* __restrict__ xrow = X + (size_t)(m0 + lrow) * CC;
  v8f c = {};
#pragma unroll
  for (int k = 0; k < CC; k += 4) {
    v2f a, b;
    a.x = xrow[k + khalf];
    a.y = xrow[k + khalf + 1];
    b.x = W[(size_t)(k + khalf) * CC + n0 + lrow];
    b.y = W[(size_t)(k + khalf + 1) * CC + n0 + lrow];
    c = __builtin_amdgcn_wmma_f32_16x16x4_f32(false, a, false, b,
                                              (short)0, c, false, false);
  }
  const int rbase = m0 + ((lane >> 4) << 3);
#pragma unroll
  for (int r = 0; r < 8; ++r)
    Y[(size_t)(rbase + r) * CC + n0 + lrow] = c[r];
}

// ---------------------------------------------------------------------------
// Per-layer init: zero the accumulation target, segment-max (encoded) and
// segment-sum buffers.
// ---------------------------------------------------------------------------
__global__ void gat_init(float* __restrict__ accum, unsigned* __restrict__ menc,
                         float* __restrict__ denom, int n_nodes) {
  const long long i = (long long)blockIdx.x * blockDim.x + threadIdx.x;
  const long long nc = (long long)n_nodes * CC;
  if (i < nc) accum[i] = 0.0f;
  if (i < (long long)n_nodes * HH) {
    menc[i] = 0u;       // encoded "very small" (below enc of any finite float)
    denom[i] = 0.0f;
  }
}

// ---------------------------------------------------------------------------
// Pass 1: logits e[E,H] = a_h . leaky_relu(xl[src] + xr[dst]) and segment max.
// One thread per (edge, head); float4 gathers stay in L2 (xl/xr are 25.6 MB).
// ---------------------------------------------------------------------------
__global__ __launch_bounds__(256) void gat_edge_logits(
    const float* __restrict__ xl, const float* __restrict__ xr,
    const float* __restrict__ att, const int* __restrict__ src,
    const int* __restrict__ dst, float* __restrict__ elog,
    unsigned* __restrict__ menc, int nedges) {
  const long long t = (long long)blockIdx.x * blockDim.x + threadIdx.x;
  if (t >= (long long)nedges * HH) return;
  const int e = (int)(t >> 2);
  const int h = (int)(t & 3);
  const int s = src[e];
  const int d = dst[e];

  const float4* __restrict__ pl = (const float4*)(xl + (size_t)s * CC + h * OO);
  const float4* __restrict__ pr = (const float4*)(xr + (size_t)d * CC + h * OO);
  const float4* __restrict__ pa = (const float4*)(att + h * OO);

  float acc = 0.0f;
#pragma unroll
  for (int j = 0; j < 4; ++j) {
    float4 l = pl[j];
    float4 r = pr[j];
    float4 av = pa[j];
    float sx = l.x + r.x; sx = sx > 0.0f ? sx : NEG_SLOPE * sx;
    float sy = l.y + r.y; sy = sy > 0.0f ? sy : NEG_SLOPE * sy;
    float sz = l.z + r.z; sz = sz > 0.0f ? sz : NEG_SLOPE * sz;
    float sw = l.w + r.w; sw = sw > 0.0f ? sw : NEG_SLOPE * sw;
    acc += sx * av.x + sy * av.y + sz * av.z + sw * av.w;
  }
  elog[t] = acc;
  atomicMax(&menc[(size_t)d * HH + h], fenc(acc));
}

// ---------------------------------------------------------------------------
// Pass 2: ex = exp(e - m[dst]); accumulate denom. elog overwritten with ex.
// ---------------------------------------------------------------------------
__global__ __launch_bounds__(256) void gat_edge_exp(
    const int* __restrict__ dst, float* __restrict__ elog,
    const unsigned* __restrict__ menc, float* __restrict__ denom, int nedges) {
  const long long t = (long long)blockIdx.x * blockDim.x + threadIdx.x;
  if (t >= (long long)nedges * HH) return;
  const int e = (int)(t >> 2);
  const int h = (int)(t & 3);
  const int d = dst[e];
  const float m = fdec(menc[(size_t)d * HH + h]);
  const float ex = __expf(elog[t] - m);
  elog[t] = ex;
  atomicAdd(&denom[(size_t)d * HH + h], ex);
}

// ---------------------------------------------------------------------------
// Pass 3: out[dst] += alpha * xl[src] (per head), via f32 atomics in L2.
// ---------------------------------------------------------------------------
__global__ __launch_bounds__(256) void gat_edge_scatter(
    const float* __restrict__ xl, const int* __restrict__ src,
    const int* __restrict__ dst, const float* __restrict__ elog,
    const float* __restrict__ denom, float* __restrict__ out, int nedges) {
  const long long t = (long long)blockIdx.x * blockDim.x + threadIdx.x;
  if (t >= (long long)nedges * HH) return;
  const int e = (int)(t >> 2);
  const int h = (int)(t & 3);
  const int s = src[e];
  const int d = dst[e];
  const float alpha = elog[t] / (denom[(size_t)d * HH + h] + 1e-16f);

  const float4* __restrict__ pl = (const float4*)(xl + (size_t)s * CC + h * OO);
  float* __restrict__ po = out + (size_t)d * CC + h * OO;
#pragma unroll
  for (int j = 0; j < 4; ++j) {
    float4 l = pl[j];
    atomicAdd(po + 4 * j + 0, l.x * alpha);
    atomicAdd(po + 4 * j + 1, l.y * alpha);
    atomicAdd(po + 4 * j + 2, l.z * alpha);
    atomicAdd(po + 4 * j + 3, l.w * alpha);
  }
}

// ---------------------------------------------------------------------------
// Bias (+ optional ReLU) applied in place on the accumulated output.
// ---------------------------------------------------------------------------
__global__ __launch_bounds__(256) void gat_bias_act(
    float* __restrict__ buf, const float* __restrict__ bias, int n_nodes,
    int do_relu) {
  const long long i = (long long)blockIdx.x * blockDim.x + threadIdx.x;
  if (i >= (long long)n_nodes * CC) return;
  float v = buf[i] + bias[i & (CC - 1)];
  if (do_relu) v = fmaxf(v, 0.0f);
  buf[i] = v;
}

// ---------------------------------------------------------------------------
// Host-side orchestration. Input order: x, edge_index, then (Wl,Wr,a,b) x3.
// ---------------------------------------------------------------------------
extern "C" void kernel_launch(void* const* d_in, const int* in_sizes, int n_in,
                              void* d_out, int out_size, void* d_ws,
                              size_t ws_size, hipStream_t stream) {
  const int N = in_sizes[0] / CC;        // 100000
  const int E = in_sizes[1] / 2;         // 1000000

  const float* x = (const float*)d_in[0];
  const int* edge = (const int*)d_in[1]; // jax default int32
  const int* src = edge;
  const int* dst = edge + E;

  const float* Wl[3] = {(const float*)d_in[2], (const float*)d_in[6],
                        (const float*)d_in[10]};
  const float* Wr[3] = {(const float*)d_in[3], (const float*)d_in[7],
                        (const float*)d_in[11]};
  const float* Av[3] = {(const float*)d_in[4], (const float*)d_in[8],
                        (const float*)d_in[12]};
  const float* Bv[3] = {(const float*)d_in[5], (const float*)d_in[9],
                        (const float*)d_in[13]};

  // workspace layout (floats)
  const size_t NC = (size_t)N * CC;      // 6.4M
  const size_t EH = (size_t)E * HH;      // 4.0M
  const size_t NH = (size_t)N * HH;      // 0.4M
  float* ws = (float*)d_ws;
  float* xl = ws;                 ws += NC;
  float* xr = ws;                 ws += NC;
  float* elog = ws;               ws += EH;
  unsigned* menc = (unsigned*)ws; ws += NH;
  float* denom = ws;              ws += NH;
  float* hA = ws;                 ws += NC;
  float* hB = ws;                 ws += NC;   // ~122 MB total

  const float* lin[3] = {x, hA, hB};
  float* lout[3] = {hA, hB, (float*)d_out};
  const int lrelu[3] = {1, 1, 0};

  const int gemm_grid = (N + 15) / 16;
  const long long nc_thr = (long long)N * CC;
  const long long eh_thr = (long long)E * HH;
  const int init_grid = (int)((nc_thr + 255) / 256);
  const int edge_grid = (int)((eh_thr + 255) / 256);

  for (int l = 0; l < 3; ++l) {
    gat_init<<<init_grid, 256, 0, stream>>>(lout[l], menc, denom, N);
    gat_gemm_wmma<<<gemm_grid, 128, 0, stream>>>(lin[l], Wl[l], xl, N);
    gat_gemm_wmma<<<gemm_grid, 128, 0, stream>>>(lin[l], Wr[l], xr, N);
    gat_edge_logits<<<edge_grid, 256, 0, stream>>>(xl, xr, Av[l], src, dst,
                                                   elog, menc, E);
    gat_edge_exp<<<edge_grid, 256, 0, stream>>>(dst, elog, menc, denom, E);
    gat_edge_scatter<<<edge_grid, 256, 0, stream>>>(xl, src, dst, elog, denom,
                                                    lout[l], E);
    gat_bias_act<<<init_grid, 256, 0, stream>>>(lout[l], Bv[l], N, lrelu[l]);
  }
}